// InterpretableBottleneck_34737695490129
// MI455X (gfx1250) — compile-verified
//
#include <hip/hip_runtime.h>

// ---------------------------------------------------------------------------
// InterpretableBottleneck forward for MI455X (gfx1250, wave32, WMMA).
// f32 WMMA (V_WMMA_F32_16X16X4_F32) everywhere: the workload is memory-bound
// on the 402MB pooling pass, so full-precision matrix ops cost nothing.
// VQ distance GEMM stages tiles in LDS via GLOBAL_LOAD_ASYNC_TO_LDS_B128
// (ASYNCcnt) and feeds WMMA from LDS.
// ---------------------------------------------------------------------------

typedef __attribute__((ext_vector_type(2))) float v2f;
typedef __attribute__((ext_vector_type(8))) float v8f;

#define D_MODEL 1024
#define SEQ     128
#define B_ROWS  256
#define H2      512
#define H       256
#define KCB     8192
#define NSPK    1000
#define POL     3
#define PROS    16
#define VIS     10

// ---------------- workspace layout (float offsets) -------------------------
#define WS_X      0u          // 256*3072
#define WS_HPRE   786432u     // 256*512
#define WS_H      917504u     // 256*512
#define WS_MPRE   1048576u    // 256*256
#define WS_CC     1114112u    // 8192
#define WS_T      1122304u    // 256*128
#define WS_BEST   1155072u    // 256 u64  (byte off 4620288, 8B aligned)
#define WS_ROWSQ  1155584u    // 256
#define WS_IDX    1155840u    // 256 int

// ---------------- output layout (float offsets) -----------------------------
#define OUT_M     0u
#define OUT_MRAW  65536u
#define OUT_LOSS  131072u
#define OUT_PERP  131073u
#define OUT_SPK   131074u
#define OUT_POL   387074u
#define OUT_PROS  387842u
#define OUT_VIS   391938u

// ---------------------------------------------------------------------------
// async copy of 16 bytes global -> LDS (ASYNCcnt-tracked, no VGPR round trip)
__device__ __forceinline__ void async_b128(void* lds, const float* g) {
    unsigned l = (unsigned)(unsigned long long)lds;   // LDS byte addr = addr[31:0]
    asm volatile("global_load_async_to_lds_b128 %0, %1, off"
                 :: "v"(l), "v"(g) : "memory");
}
__device__ __forceinline__ void wait_async0() {
    asm volatile("s_wait_asynccnt 0" ::: "memory");
}

// ---------------------------------------------------------------------------
// 1) mean-pool over sequence dim; x[b, 0:1024|1024:2048|2048:3072]
__global__ void ib_pool_kernel(const float* __restrict__ ut,
                               const float* __restrict__ ua,
                               const float* __restrict__ uv,
                               float* __restrict__ x) {
    int i = blockIdx.x * blockDim.x + threadIdx.x;
    if (i >= B_ROWS * 3 * D_MODEL) return;
    int b = i / (3 * D_MODEL);
    int d = i % (3 * D_MODEL);
    const float* src = (d < D_MODEL) ? ut : ((d < 2 * D_MODEL) ? ua : uv);
    int dd = d % D_MODEL;
    const float* p = src + (size_t)b * SEQ * D_MODEL + dd;
    float s = 0.f;
#pragma unroll 4
    for (int t = 0; t < SEQ; ++t) s += p[(size_t)t * D_MODEL];
    x[i] = s * (1.0f / SEQ);
}

// ---------------------------------------------------------------------------
// Generic f32 WMMA GEMM: C[M,N] = act(A[M,K] @ B[K,N] + bias), one wave per
// 16x16 output tile. Ragged N handled by clamping the B column (branch-free
// inner loop; padded lanes compute garbage that is never stored).
__global__ void ib_wmma_gemm(const float* __restrict__ A,
                             const float* __restrict__ Bm,
                             const float* __restrict__ bias,
                             float* __restrict__ C,
                             int M, int N, int K, int relu) {
    int lane = threadIdx.x;           // blockDim.x == 32 (one wave, EXEC all 1s)
    int half = lane >> 4;
    int r    = lane & 15;
    int row0 = blockIdx.y * 16;
    int col0 = blockIdx.x * 16;
    int col  = col0 + r;
    int colc = (col < N) ? col : (N - 1);   // clamp: keeps loop branch-free

    const float* Arow = A + (size_t)(row0 + r) * K + 2 * half;
    const float* Bcol = Bm + (size_t)(2 * half) * N + colc;
    v8f acc = {};
#pragma unroll 4
    for (int k = 0; k < K; k += 4) {
        v2f a, b;
        a.x = Arow[k];
        a.y = Arow[k + 1];
        b.x = Bcol[(size_t)k * N];
        b.y = Bcol[(size_t)k * N + N];
        acc = __builtin_amdgcn_wmma_f32_16x16x4_f32(
            false, a, false, b, (short)0, acc, false, false);
    }
    if (col < N) {
        float bv = bias[col];
#pragma unroll
        for (int j = 0; j < 8; ++j) {
            float v = acc[j] + bv;
            if (relu) v = fmaxf(v, 0.0f);
            C[(size_t)(row0 + half * 8 + j) * N + col] = v;
        }
    }
}

// ---------------------------------------------------------------------------
// Row-wise LayerNorm; one 256-thread block per row.
__global__ void ib_ln_kernel(const float* __restrict__ X,
                             const float* __restrict__ g,
                             const float* __restrict__ b,
                             float* __restrict__ Y, int cols) {
    __shared__ float sbuf[256];
    int row = blockIdx.x, t = threadIdx.x;
    const float* xr = X + (size_t)row * cols;
    float s = 0.f;
    for (int c = t; c < cols; c += 256) s += xr[c];
    sbuf[t] = s; __syncthreads();
    for (int o = 128; o > 0; o >>= 1) { if (t < o) sbuf[t] += sbuf[t + o]; __syncthreads(); }
    float mu = sbuf[0] / cols; __syncthreads();
    float v = 0.f;
    for (int c = t; c < cols; c += 256) { float d = xr[c] - mu; v += d * d; }
    sbuf[t] = v; __syncthreads();
    for (int o = 128; o > 0; o >>= 1) { if (t < o) sbuf[t] += sbuf[t + o]; __syncthreads(); }
    float rinv = rsqrtf(sbuf[0] / cols + 1e-5f);
    for (int c = t; c < cols; c += 256)
        Y[(size_t)row * cols + c] = (xr[c] - mu) * rinv * g[c] + b[c];
}

// ---------------------------------------------------------------------------
// per-codebook-row squared norms
__global__ void ib_cnorm_kernel(const float* __restrict__ cb, float* __restrict__ cc) {
    __shared__ float sbuf[256];
    int k = blockIdx.x, t = threadIdx.x;
    float v = cb[(size_t)k * H + t];
    sbuf[t] = v * v; __syncthreads();
    for (int o = 128; o > 0; o >>= 1) { if (t < o) sbuf[t] += sbuf[t + o]; __syncthreads(); }
    if (t == 0) cc[k] = sbuf[0];
}

__global__ void ib_init_kernel(unsigned long long* __restrict__ best) {
    int i = threadIdx.x;
    if (i < B_ROWS) best[i] = ~0ull;
}

// ---------------------------------------------------------------------------
// VQ: score[b,k] = ||c_k||^2 - 2 m_b . c_k. Block = 64 threads (2 waves),
// tile = 16 rows x 32 codebook cols. m-tile and codebook-tile staged in LDS
// via GLOBAL_LOAD_ASYNC_TO_LDS_B128, WMMA fed from LDS. Per-row argmin via
// lane butterfly + GLOBAL_ATOMIC_MIN_U64 on packed (mapped_score, index).
#define VQ_COLS 32
__global__ void ib_vq_kernel(const float* __restrict__ m,
                             const float* __restrict__ cb,
                             const float* __restrict__ cc,
                             unsigned long long* __restrict__ best) {
    __shared__ float smA[16 * H];        // 16 KB
    __shared__ float smB[VQ_COLS * H];   // 32 KB
    int t    = threadIdx.x;              // 0..63
    int lane = t & 31;
    int wv   = t >> 5;                   // wave id 0..1
    int half = lane >> 4;
    int r    = lane & 15;
    int row0 = blockIdx.y * 16;
    int col0 = blockIdx.x * VQ_COLS;

    const float* gA = m  + (size_t)row0 * H;   // 16 contiguous rows
    const float* gB = cb + (size_t)col0 * H;   // 32 contiguous codebook rows

    for (int i = t; i < (16 * H) / 4; i += 64)
        async_b128(&smA[i * 4], gA + i * 4);
    for (int i = t; i < (VQ_COLS * H) / 4; i += 64)
        async_b128(&smB[i * 4], gB + i * 4);
    wait_async0();
    __syncthreads();

    int cl  = wv * 16 + r;               // local codebook col for this lane
    int col = col0 + cl;
    const float* Ar = &smA[r  * H + 2 * half];
    const float* Br = &smB[cl * H + 2 * half];
    v8f acc = {};
#pragma unroll 8
    for (int k = 0; k < H; k += 4) {
        v2f a, b;
        a.x = Ar[k]; a.y = Ar[k + 1];
        b.x = Br[k]; b.y = Br[k + 1];
        acc = __builtin_amdgcn_wmma_f32_16x16x4_f32(
            false, a, false, b, (short)0, acc, false, false);
    }
    float ccv = cc[col];
#pragma unroll
    for (int j = 0; j < 8; ++j) {
        float score = ccv - 2.0f * acc[j];
        unsigned u = __float_as_uint(score);
        u = (u & 0x80000000u) ? ~u : (u | 0x80000000u);   // order-preserving map
        unsigned long long pk = ((unsigned long long)u << 32) | (unsigned)col;
        for (int msk = 1; msk < 16; msk <<= 1) {          // min over 16-lane group
            unsigned lo = (unsigned)pk, hi = (unsigned)(pk >> 32);
            unsigned olo = __shfl_xor(lo, msk, 32);
            unsigned ohi = __shfl_xor(hi, msk, 32);
            unsigned long long other = ((unsigned long long)ohi << 32) | olo;
            if (other < pk) pk = other;
        }
        if (r == 0)
            atomicMin(&best[row0 + half * 8 + j], pk);    // GLOBAL_ATOMIC_MIN_U64
    }
}

// ---------------------------------------------------------------------------
// gather quantized rows, per-row squared error, record indices
__global__ void ib_gather_kernel(const unsigned long long* __restrict__ best,
                                 const float* __restrict__ cb,
                                 const float* __restrict__ m,
                                 float* __restrict__ Mout,
                                 int* __restrict__ idx_out,
                                 float* __restrict__ row_sq) {
    __shared__ float sbuf[256];
    int b = blockIdx.x, d = threadIdx.x;
    int k = (int)(best[b] & 0xFFFFFFFFull);
    float q  = cb[(size_t)k * H + d];
    float mv = m[(size_t)b * H + d];
    Mout[(size_t)b * H + d] = q;          // forward M == quantized
    float diff = q - mv;
    sbuf[d] = diff * diff; __syncthreads();
    for (int o = 128; o > 0; o >>= 1) { if (d < o) sbuf[d] += sbuf[d + o]; __syncthreads(); }
    if (d == 0) { row_sq[b] = sbuf[0]; idx_out[b] = k; }
}

// ---------------------------------------------------------------------------
// deterministic loss + perplexity (8192-bin histogram in LDS)
__global__ void ib_finalize_kernel(const float* __restrict__ row_sq,
                                   const int* __restrict__ idx,
                                   float* __restrict__ out_loss,
                                   float* __restrict__ out_perp) {
    __shared__ int   hist[KCB];   // 32 KB of the 320 KB WGP LDS
    __shared__ float sb[256];
    int t = threadIdx.x;
    for (int i = t; i < KCB; i += 256) hist[i] = 0;
    __syncthreads();
    if (t < B_ROWS) atomicAdd(&hist[idx[t]], 1);
    __syncthreads();
    float s = 0.f;
    for (int i = t; i < KCB; i += 256) {
        int c = hist[i];
        if (c) { float p = c * (1.0f / B_ROWS); s += p * logf(p + 1e-10f); }
    }
    sb[t] = s; __syncthreads();
    for (int o = 128; o > 0; o >>= 1) { if (t < o) sb[t] += sb[t + o]; __syncthreads(); }
    float ent = sb[0];
    __syncthreads();
    sb[t] = row_sq[t]; __syncthreads();
    for (int o = 128; o > 0; o >>= 1) { if (t < o) sb[t] += sb[t + o]; __syncthreads(); }
    if (t == 0) {
        // vq_loss = q_latent + 0.25*e_latent = 1.25 * mean((q-m)^2)
        out_loss[0] = sb[0] * (1.25f / (float)(B_ROWS * H));
        out_perp[0] = expf(-ent) / (float)KCB;
    }
}

// ---------------------------------------------------------------------------
// small heads (pol/pros/vis): one thread per output element, K=256 dot
__global__ void ib_heads_kernel(const float* __restrict__ Mout,
                                const float* __restrict__ pW, const float* __restrict__ pb,
                                const float* __restrict__ prW, const float* __restrict__ prb,
                                const float* __restrict__ vW, const float* __restrict__ vb,
                                float* __restrict__ out_pol,
                                float* __restrict__ out_pro,
                                float* __restrict__ out_vis) {
    int i = blockIdx.x * blockDim.x + threadIdx.x;
    const int per_row = POL + PROS + VIS;   // 29
    if (i >= B_ROWS * per_row) return;
    int b = i / per_row;
    int j = i % per_row;
    const float *W, *bias; float* dst; int n;
    if (j < POL)            { W = pW;  bias = pb;  dst = out_pol; n = POL; }
    else if (j < POL + PROS){ j -= POL; W = prW; bias = prb; dst = out_pro; n = PROS; }
    else                    { j -= POL + PROS; W = vW; bias = vb; dst = out_vis; n = VIS; }
    float s = bias[j];
    const float* mr = Mout + (size_t)b * H;
#pragma unroll 4
    for (int k = 0; k < H; ++k) s += mr[k] * W[(size_t)k * n + j];
    dst[(size_t)b * n + j] = s;
}

// ---------------------------------------------------------------------------
extern "C" void kernel_launch(void* const* d_in, const int* in_sizes, int n_in,
                              void* d_out, int out_size, void* d_ws, size_t ws_size,
                              hipStream_t stream) {
    (void)in_sizes; (void)n_in; (void)out_size; (void)ws_size;
    const float* u_t   = (const float*)d_in[0];
    const float* u_a   = (const float*)d_in[1];
    const float* u_v   = (const float*)d_in[2];
    const float* W1    = (const float*)d_in[3];
    const float* b1    = (const float*)d_in[4];
    const float* ln1_g = (const float*)d_in[5];
    const float* ln1_b = (const float*)d_in[6];
    const float* W2    = (const float*)d_in[7];
    const float* b2    = (const float*)d_in[8];
    const float* ln2_g = (const float*)d_in[9];
    const float* ln2_b = (const float*)d_in[10];
    const float* cb    = (const float*)d_in[11];
    const float* dW1   = (const float*)d_in[12];
    const float* db1   = (const float*)d_in[13];
    const float* dW2   = (const float*)d_in[14];
    const float* db2   = (const float*)d_in[15];
    const float* pW    = (const float*)d_in[16];
    const float* pb    = (const float*)d_in[17];
    const float* prW   = (const float*)d_in[18];
    const float* prb   = (const float*)d_in[19];
    const float* vW    = (const float*)d_in[20];
    const float* vb    = (const float*)d_in[21];

    float* ws = (float*)d_ws;
    float* o  = (float*)d_out;

    float* x     = ws + WS_X;
    float* hpre  = ws + WS_HPRE;
    float* h     = ws + WS_H;
    float* mpre  = ws + WS_MPRE;
    float* cc    = ws + WS_CC;
    float* tbuf  = ws + WS_T;
    unsigned long long* best = (unsigned long long*)(ws + WS_BEST);
    float* row_sq = ws + WS_ROWSQ;
    int*   idx    = (int*)(ws + WS_IDX);

    float* m_raw = o + OUT_MRAW;

    // 1) mean-pool -> x[256,3072]
    ib_pool_kernel<<<(B_ROWS * 3 * D_MODEL + 255) / 256, 256, 0, stream>>>(u_t, u_a, u_v, x);
    // 2) h = LN(relu(x@W1 + b1))
    ib_wmma_gemm<<<dim3(H2 / 16, B_ROWS / 16), 32, 0, stream>>>(x, W1, b1, hpre,
                                                               B_ROWS, H2, 3 * D_MODEL, 1);
    ib_ln_kernel<<<B_ROWS, 256, 0, stream>>>(hpre, ln1_g, ln1_b, h, H2);
    // 3) m_raw = LN(h@W2 + b2)
    ib_wmma_gemm<<<dim3(H / 16, B_ROWS / 16), 32, 0, stream>>>(h, W2, b2, mpre,
                                                              B_ROWS, H, H2, 0);
    ib_ln_kernel<<<B_ROWS, 256, 0, stream>>>(mpre, ln2_g, ln2_b, m_raw, H);
    // 4) VQ
    ib_cnorm_kernel<<<KCB, 256, 0, stream>>>(cb, cc);
    ib_init_kernel<<<1, 256, 0, stream>>>(best);
    ib_vq_kernel<<<dim3(KCB / VQ_COLS, B_ROWS / 16), 64, 0, stream>>>(m_raw, cb, cc, best);
    ib_gather_kernel<<<B_ROWS, 256, 0, stream>>>(best, cb, m_raw, o + OUT_M, idx, row_sq);
    ib_finalize_kernel<<<1, 256, 0, stream>>>(row_sq, idx, o + OUT_LOSS, o + OUT_PERP);
    // 5) speaker head: relu(M@dW1+db1)@dW2+db2
    ib_wmma_gemm<<<dim3(128 / 16, B_ROWS / 16), 32, 0, stream>>>(o + OUT_M, dW1, db1, tbuf,
                                                                B_ROWS, 128, H, 1);
    ib_wmma_gemm<<<dim3((NSPK + 15) / 16, B_ROWS / 16), 32, 0, stream>>>(tbuf, dW2, db2,
                                                                        o + OUT_SPK,
                                                                        B_ROWS, NSPK, 128, 0);
    // 6) tiny heads
    ib_heads_kernel<<<(B_ROWS * (POL + PROS + VIS) + 255) / 256, 256, 0, stream>>>(
        o + OUT_M, pW, pb, prW, prb, vW, vb,
        o + OUT_POL, o + OUT_PROS, o + OUT_VIS);
}